// Network1_53180285059450
// MI455X (gfx1250) — compile-verified
//
#include <hip/hip_runtime.h>
#include <hip/hip_bf16.h>

// ---------------------------------------------------------------------------
// SLAYER spiking CNN forward for MI455X (gfx1250, wave32, WMMA).
// Layouts:  G = (B,C,H,W,T)  T-contiguous (temporal ops / final output)
//           P = (B,T,C,H,W)  per-timestep images (spatial convs)
// Pipeline:
//   psp0          : spikeInput(G) --temporal conv K=8--> psp1(P)        [LDS transpose]
//   conv<2,5,2>   : psp1 -> u1(P)                                       [fp32 WMMA]
//   scan+psp<8,16>: u1 -> p1(P)   (LIF scan theta=30 fused w/ next PSP)
//   conv<8,3,1>   : p1 -> u2 ; scan+psp<16,16>(theta=50) -> p2
//   conv<8,3,1>   : p2 -> u3 ; scan+psp<16,31>(theta=50) -> p3
//   final         : convT(2x2,s2) + bilinear2x(psp1) + LIF(theta=100,K=31) -> d_out(G)
// ---------------------------------------------------------------------------

typedef __attribute__((ext_vector_type(2))) float v2f;
typedef __attribute__((ext_vector_type(8))) float v8f;

// ---------------------------------------------------------------------------
// Kernel 1: temporal PSP conv of the raw spike input (tau=1, K=8 taps),
// transposing G->P via LDS.  One block per (b,c,y): 64 x-rows * 128 t.
// ---------------------------------------------------------------------------
__global__ __launch_bounds__(256)
void psp0_kernel(const float* __restrict__ x, float* __restrict__ p)
{
    constexpr int Cc = 2, W = 64, T = 128, K = 8;
    __shared__ float sIn [64][T + 1];   // +1 pad: bank-conflict-free column reads
    __shared__ float sOut[64][T + 1];

    const int blk = blockIdx.x;            // (b*2+c)*64 + y
    const int y   = blk & 63;
    const int bc  = blk >> 6;              // b*2+c
    const int b   = bc >> 1, c = bc & 1;
    const int tid = threadIdx.x;

    const float* src = x + ((size_t)bc * 64 + y) * (size_t)(W * T); // contiguous 32KB
    for (int fi = tid; fi < W * T; fi += 256)
        sIn[fi >> 7][fi & 127] = src[fi];

    float kc[K];
#pragma unroll
    for (int j = 0; j < K; ++j)            // alpha kernel, tau=1, mult=1
        kc[j] = (float)j * __expf(1.f - (float)j);
    __syncthreads();

    {   // 4 threads per x-row, 32 t each
        const int xr = tid >> 2;
        const int t0 = (tid & 3) * 32;
        for (int t = t0; t < t0 + 32; ++t) {
            float acc = 0.f;
#pragma unroll
            for (int j = 0; j < K; ++j) {
                int ts = t - j;
                float v = (ts >= 0) ? sIn[xr][ts] : 0.f;
                acc = fmaf(kc[j], v, acc);
            }
            sOut[xr][t] = acc;
        }
    }
    __syncthreads();

    // coalesced transposed store to P layout: p[((b*T+t)*2+c)*4096 + y*64 + x]
    const size_t pbase = ((size_t)b * T * Cc + c) * 4096u + (size_t)y * W;
    for (int fi = tid; fi < W * T; fi += 256) {
        int t  = fi >> 6;
        int xw = fi & 63;
        p[pbase + (size_t)t * (Cc * 4096) + xw] = sOut[xw][t];
    }
}

// ---------------------------------------------------------------------------
// Kernel 2: per-timestep 2D conv as implicit GEMM on fp32 WMMA 16x16x4.
//   M = out-channel (8, padded to 16), N = 16 x-positions, K = 4-chunk over
//   the (tap-major, ic-minor) im2col reduction (zero padded to mult of 4).
// Block = one (b,t) image * 8 output rows; 8 waves, wave w -> row w, 4 x-tiles.
// ---------------------------------------------------------------------------
template <int CIN, int KSZ, int PAD>
__global__ __launch_bounds__(256)
void conv_wmma_kernel(const float* __restrict__ in, const float* __restrict__ w,
                      float* __restrict__ out)
{
    constexpr int COUT = 8;
    constexpr int H = 64, W = 64, HW = 4096;
    constexpr int YT = 8;
    constexpr int R  = KSZ * KSZ * CIN;
    constexpr int RP = (R + 3) & ~3;
    constexpr int TILE_H = YT + KSZ - 1;
    constexpr int TILE_W = W + 2 * PAD;

    __shared__ float wl[RP][16];
    __shared__ float il[CIN][TILE_H][TILE_W];

    const int bt  = blockIdx.x;            // b*T + t
    const int y0  = blockIdx.y * YT;
    const int tid = threadIdx.x;

    // weights -> LDS, layout [reduction r][oc16], zero padded
    for (int fi = tid; fi < RP * 16; fi += 256) {
        int r = fi >> 4, oc = fi & 15;
        float v = 0.f;
        if (r < R && oc < COUT) {
            int ic = r % CIN, tap = r / CIN;
            int ky = tap / KSZ, kx = tap % KSZ;
            v = w[((oc * CIN + ic) * KSZ + ky) * KSZ + kx];
        }
        wl[r][oc] = v;
    }
    // input tile + zero halo -> LDS (coalesced row loads)
    const float* ib = in + (size_t)bt * CIN * HW;
    for (int fi = tid; fi < CIN * TILE_H * TILE_W; fi += 256) {
        int col = fi % TILE_W;
        int row = fi / TILE_W;
        int yy  = row % TILE_H, ic = row / TILE_H;
        int gy  = y0 - PAD + yy, gx = col - PAD;
        float v = 0.f;
        if (gy >= 0 && gy < H && gx >= 0 && gx < W)
            v = ib[ic * HW + gy * W + gx];
        il[ic][yy][col] = v;
    }
    __syncthreads();

    const int wave = tid >> 5;
    const int lane = tid & 31;
    const int kh   = lane >> 4;            // 0 -> K 0,1 ; 1 -> K 2,3
    const int m16  = lane & 15;            // M(oc) for A, N(x) for B

    const int ly = wave;                   // local row 0..7
    float* ob = out + (size_t)bt * COUT * HW + (size_t)(y0 + ly) * W;

    for (int xt = 0; xt < 4; ++xt) {
        const int x0 = xt * 16;
        v8f acc = {};
#pragma unroll
        for (int c = 0; c < RP / 4; ++c) {
            const int k0 = c * 4 + kh * 2;
            v2f a, bv;
            a.x = wl[k0][m16];
            a.y = wl[k0 + 1][m16];
            float b0 = 0.f, b1 = 0.f;
            {
                int r = k0;
                if (r < R) {
                    int ic = r % CIN, tap = r / CIN;
                    int ky = tap / KSZ, kx = tap % KSZ;
                    b0 = il[ic][ly + ky][x0 + m16 + kx];
                }
                r = k0 + 1;
                if (r < R) {
                    int ic = r % CIN, tap = r / CIN;
                    int ky = tap / KSZ, kx = tap % KSZ;
                    b1 = il[ic][ly + ky][x0 + m16 + kx];
                }
            }
            bv.x = b0; bv.y = b1;
            acc = __builtin_amdgcn_wmma_f32_16x16x4_f32(
                      false, a, false, bv, (short)0, acc, false, false);
        }
        // D vgpr r, lanes 0-15: (oc=r, x=x0+lane); lanes 16-31 hold padded oc>=8
        if (lane < 16) {
#pragma unroll
            for (int r = 0; r < 8; ++r)
                ob[r * HW + x0 + lane] = acc[r];
        }
    }
}

// ---------------------------------------------------------------------------
// Kernel 3: fused LIF scan (refractory feedback) + next-layer temporal PSP.
// Thread owns one (b,c,y,x); both buffers update with one FMA per slot per
// step (no register shifting).  u read / p written coalesced in P layout.
//   LR = KRef-1 refractory slots (coef ref[i+1]);  LP = Ksrm-1 psp slots.
// ---------------------------------------------------------------------------
template <int LR, int LP>
__global__ __launch_bounds__(256)
void scan_psp_kernel(const float* __restrict__ u, float* __restrict__ p,
                     float theta, float tauR, float multR, float tauS)
{
    constexpr int T = 128, C = 8, HW = 4096, CHW = C * HW;
    const int gid = blockIdx.x * 256 + threadIdx.x;
    const int yx  = gid & (HW - 1);
    const int bc  = gid >> 12;
    const int b   = bc >> 3, c = bc & 7;
    const size_t base = ((size_t)b * T * C + c) * HW + yx;

    float refc[LR], kc[LP];
#pragma unroll
    for (int i = 0; i < LR; ++i) {
        float j = (float)(i + 1);
        refc[i] = multR * (j / tauR) * __expf(1.f - j / tauR);
    }
#pragma unroll
    for (int i = 0; i < LP; ++i) {
        float j = (float)(i + 1);
        kc[i] = (j / tauS) * __expf(1.f - j / tauS);
    }
    float rb[LR], pb[LP];
#pragma unroll
    for (int i = 0; i < LR; ++i) rb[i] = 0.f;
#pragma unroll
    for (int i = 0; i < LP; ++i) pb[i] = 0.f;

    for (int t = 0; t < T; ++t) {
        float ut   = u[base + (size_t)t * CHW];
        float ueff = ut + rb[0];
        float s    = (ueff >= theta) ? 1.0f : 0.0f;
        p[base + (size_t)t * CHW] = pb[0];      // srm[0]==0 -> p_t = pb[0]
#pragma unroll
        for (int i = 0; i < LR - 1; ++i) rb[i] = fmaf(s, refc[i], rb[i + 1]);
        rb[LR - 1] = s * refc[LR - 1];
#pragma unroll
        for (int i = 0; i < LP - 1; ++i) pb[i] = fmaf(s, kc[i], pb[i + 1]);
        pb[LP - 1] = s * kc[LP - 1];
    }
}

// ---------------------------------------------------------------------------
// Kernel 4: fused  u4 = convT(2x2,s2)(p3) + bilinear2x(psp1)  + LIF scan
// (theta=100, KRef=31), writing s4 straight to d_out in (B,2,128,128,T).
// ---------------------------------------------------------------------------
template <int LR>
__global__ __launch_bounds__(256)
void final_kernel(const float* __restrict__ p3, const float* __restrict__ psp1,
                  const float* __restrict__ w4, float* __restrict__ out)
{
    constexpr int T = 128, HO = 128, WO = 128, H = 64, W = 64;
    constexpr int C3 = 8, C1 = 2;
    const int gid = blockIdx.x * 256 + threadIdx.x;  // (b, oc, Y, X)
    const int X  = gid & 127;
    const int Y  = (gid >> 7) & 127;
    const int oc = (gid >> 14) & 1;
    const int b  = gid >> 15;

    const int y = Y >> 1, x = X >> 1;
    const int dy = Y & 1, dx = X & 1;

    float wv[8];                        // convT weights for (oc,dy,dx)
#pragma unroll
    for (int ic = 0; ic < 8; ++ic)
        wv[ic] = w4[((oc * 8 + ic) * 2 + dy) * 2 + dx];

    // bilinear 2x taps, align_corners=False, edge clamp
    int ya, yb, xa, xb; float wya, wyb, wxa, wxb;
    if (dy == 0) { ya = (y > 0) ? y - 1 : 0; yb = y; wya = 0.25f; wyb = 0.75f; }
    else         { ya = y; yb = (y < H - 1) ? y + 1 : H - 1; wya = 0.75f; wyb = 0.25f; }
    if (dx == 0) { xa = (x > 0) ? x - 1 : 0; xb = x; wxa = 0.25f; wxb = 0.75f; }
    else         { xa = x; xb = (x < W - 1) ? x + 1 : W - 1; wxa = 0.75f; wxb = 0.25f; }

    const size_t p3base = (size_t)b * T * C3 * 4096u + (size_t)y * W + x;
    const size_t p1base = (size_t)b * T * C1 * 4096u + (size_t)oc * 4096u;
    const int oAA = ya * W + xa, oAB = ya * W + xb;
    const int oBA = yb * W + xa, oBB = yb * W + xb;

    float refc[LR], rb[LR];
#pragma unroll
    for (int i = 0; i < LR; ++i) {      // theta=100, tauRef=4 -> mult=-200
        float j = (float)(i + 1);
        refc[i] = -200.f * (j * 0.25f) * __expf(1.f - j * 0.25f);
    }
#pragma unroll
    for (int i = 0; i < LR; ++i) rb[i] = 0.f;

    float* ob = out + (((size_t)(b * 2 + oc) * HO + Y) * WO + X) * T;

    for (int t4 = 0; t4 < T; t4 += 4) {
        float svv[4];
#pragma unroll
        for (int q = 0; q < 4; ++q) {
            const int t = t4 + q;
            const float* pp3 = p3 + p3base + (size_t)t * (C3 * 4096);
            float u = 0.f;
#pragma unroll
            for (int ic = 0; ic < 8; ++ic)
                u = fmaf(wv[ic], pp3[ic * 4096], u);
            const float* pp1 = psp1 + p1base + (size_t)t * (C1 * 4096);
            u += wya * (wxa * pp1[oAA] + wxb * pp1[oAB])
               + wyb * (wxa * pp1[oBA] + wxb * pp1[oBB]);
            float ueff = u + rb[0];
            float s = (ueff >= 100.f) ? 1.f : 0.f;
#pragma unroll
            for (int i = 0; i < LR - 1; ++i) rb[i] = fmaf(s, refc[i], rb[i + 1]);
            rb[LR - 1] = s * refc[LR - 1];
            svv[q] = s;
        }
        float4 v; v.x = svv[0]; v.y = svv[1]; v.z = svv[2]; v.w = svv[3];
        *(float4*)(ob + t4) = v;
    }
}

// ---------------------------------------------------------------------------
extern "C" void kernel_launch(void* const* d_in, const int* in_sizes, int n_in,
                              void* d_out, int out_size, void* d_ws, size_t ws_size,
                              hipStream_t stream)
{
    (void)in_sizes; (void)n_in; (void)out_size;
    const float* spikeIn = (const float*)d_in[0];   // (8,2,64,64,128)
    const float* w1 = (const float*)d_in[1];        // (8,2,5,5)
    const float* w2 = (const float*)d_in[2];        // (8,8,3,3)
    const float* w3 = (const float*)d_in[3];        // (8,8,3,3)
    const float* w4 = (const float*)d_in[4];        // (2,8,2,2)
    float* out = (float*)d_out;                     // (8,2,128,128,128)

    // workspace partition (needs ~302 MB): psp1 | bufU | bufP
    float* psp1 = (float*)d_ws;                     //  8*128*2*4096 floats (33.5MB)
    float* bufU = psp1 + (size_t)8 * 128 * 2 * 4096;//  8*128*8*4096 floats (134MB)
    float* bufP = bufU + (size_t)8 * 128 * 8 * 4096;//  8*128*8*4096 floats (134MB)
    (void)ws_size;

    // 1) psp1 = psp(spikeInput, srm tau=1, K=8), G->P transpose
    psp0_kernel<<<dim3(8 * 2 * 64), 256, 0, stream>>>(spikeIn, psp1);

    // 2) u1 = conv 5x5 (2->8, pad 2)
    conv_wmma_kernel<2, 5, 2><<<dim3(1024, 8), 256, 0, stream>>>(psp1, w1, bufU);
    // 3) s1 = LIF(theta=30, tauRef=1, K=8) ; p1 = psp(s1, tau=2, K=16)
    scan_psp_kernel<7, 15><<<dim3(1024), 256, 0, stream>>>(bufU, bufP, 30.f, 1.f, -60.f, 2.f);

    // 4) u2 = conv 3x3 (8->8, pad 1)
    conv_wmma_kernel<8, 3, 1><<<dim3(1024, 8), 256, 0, stream>>>(bufP, w2, bufU);
    // 5) s2 = LIF(theta=50, tauRef=2, K=16) ; p2 = psp(s2, tau=2, K=16)
    scan_psp_kernel<15, 15><<<dim3(1024), 256, 0, stream>>>(bufU, bufP, 50.f, 2.f, -100.f, 2.f);

    // 6) u3 = conv 3x3 (8->8, pad 1)
    conv_wmma_kernel<8, 3, 1><<<dim3(1024, 8), 256, 0, stream>>>(bufP, w3, bufU);
    // 7) s3 = LIF(theta=50, tauRef=2, K=16) ; p3 = psp(s3, tau=4, K=31)
    scan_psp_kernel<15, 30><<<dim3(1024), 256, 0, stream>>>(bufU, bufP, 50.f, 2.f, -100.f, 4.f);

    // 8) s4 = LIF(convT(p3) + bilinear2x(psp1), theta=100, tauRef=4, K=31)
    final_kernel<30><<<dim3(1024), 256, 0, stream>>>(bufP, psp1, w4, out);
}